// FidelityModelWithSAE_13383118094459
// MI455X (gfx1250) — compile-verified
//
#include <hip/hip_runtime.h>
#include <stdint.h>

// energy + per-molecule segment-sum of sae_tensor[numbers + 200]  (mol_idx sorted)
#define FID_SHIFT 200   // FIDELITY_LEVEL * FIDELITY_OFFSET

typedef __attribute__((ext_vector_type(2))) float v2f;
typedef __attribute__((ext_vector_type(8))) float v8f;

__global__ __launch_bounds__(256) void sae_segsum_kernel(
    const float* __restrict__ energy,
    const float* __restrict__ sae,
    const int*   __restrict__ numbers,
    const int*   __restrict__ mol_idx,
    float*       __restrict__ out,
    int n_atoms, int n_mol, int sae_n)
{
    __shared__ float s_sae[640];  // 518 used

    // Stage SAE table into LDS via CDNA5 async global->LDS DMA path.
    for (int i = (int)threadIdx.x; i < sae_n; i += (int)blockDim.x) {
        uint32_t lds_addr = (uint32_t)(uintptr_t)&s_sae[i];
        uint32_t goff     = (uint32_t)i * 4u;
        asm volatile("global_load_async_to_lds_b32 %0, %1, %2"
                     :: "v"(lds_addr), "v"(goff), "s"(sae) : "memory");
    }
    asm volatile("s_wait_asynccnt 0" ::: "memory");
    __syncthreads();

    const int lane = (int)(threadIdx.x & 31u);
    const int m    = (int)(blockIdx.x * (blockDim.x >> 5) + (threadIdx.x >> 5)); // wave id
    if (m >= n_mol) return;  // wave-uniform: EXEC stays all-1s for live waves

    // mol_idx is sorted: lanes 0-15 lower_bound(m), lanes 16-31 lower_bound(m+1).
    int target = m + (lane >= 16 ? 1 : 0);
    int lo = 0, hi = n_atoms;
    while (lo < hi) {
        int mid = (lo + hi) >> 1;
        if (mol_idx[mid] < target) lo = mid + 1; else hi = mid;
    }
    const int beg = __shfl(lo, 0);
    const int end = __shfl(lo, 16);

    // Coalesced streaming of this molecule's contiguous atom run.
    float acc = 0.0f;
    for (int i = beg + lane; i < end; i += 32)
        acc += s_sae[numbers[i] + FID_SHIFT];

    // 32-lane f32 reduction. WMMA f32 16x16x4 with B=ones:
    //   A[r][0]=acc(lane r), A[r][1]=0, A[r][2]=acc(lane r+16), A[r][3]=0
    //   => D[M][*] = acc[M] + acc[M+16]; each lane holds 8 distinct row-sums,
    //   lanes 0-15 rows 0-7, lanes 16-31 rows 8-15. Sum 8 VGPRs + one xor-16.
    float total;
#if __has_builtin(__builtin_amdgcn_wmma_f32_16x16x4_f32)
    v2f a; a[0] = acc;  a[1] = 0.0f;
    v2f b; b[0] = 1.0f; b[1] = 1.0f;
    v8f c = {};
    v8f d = __builtin_amdgcn_wmma_f32_16x16x4_f32(
        /*neg_a=*/false, a, /*neg_b=*/false, b,
        /*c_mod=*/(short)0, c, /*reuse_a=*/false, /*reuse_b=*/false);
    float part = ((d[0] + d[1]) + (d[2] + d[3])) + ((d[4] + d[5]) + (d[6] + d[7]));
    total = part + __shfl_xor(part, 16);
#else
    total = acc;
    #pragma unroll
    for (int off = 16; off > 0; off >>= 1) total += __shfl_xor(total, off);
#endif

    if (lane == 0) out[m] = energy[m] + total;
}

extern "C" void kernel_launch(void* const* d_in, const int* in_sizes, int n_in,
                              void* d_out, int out_size, void* d_ws, size_t ws_size,
                              hipStream_t stream) {
    const float* energy  = (const float*)d_in[0];
    const float* sae     = (const float*)d_in[1];
    const int*   numbers = (const int*)d_in[2];
    const int*   mol_idx = (const int*)d_in[3];
    float* out = (float*)d_out;

    const int n_mol   = in_sizes[0];
    const int sae_n   = in_sizes[1];
    const int n_atoms = in_sizes[2];

    const int threads = 256;              // 8 wave32 per block
    const int wpb     = threads / 32;     // one wave per molecule
    const int blocks  = (n_mol + wpb - 1) / wpb;

    sae_segsum_kernel<<<blocks, threads, 0, stream>>>(
        energy, sae, numbers, mol_idx, out, n_atoms, n_mol, sae_n);
}